// EMA_34136400069276
// MI455X (gfx1250) — compile-verified
//
#include <hip/hip_runtime.h>

typedef __attribute__((ext_vector_type(2))) float v2f;
typedef __attribute__((ext_vector_type(8))) float v8f;

#define GN_EPS 1e-5f

__device__ __forceinline__ float sigmoidf_(float v) { return 1.f / (1.f + __expf(-v)); }

// One block per bg-group (512 total). 1024 threads = 32 wave32.
// Dynamic LDS layout (floats):
//   simg : [0,      65536)  16ch x 64 x 64 image (scaled in place before async store-out)
//   wmap : [65536,  69632)  weights map (reused as GN-reduction scratch)
//   a_h  : [69632,  70656)  16 x 64 row gates (later scaled by kA)
//   a_w  : [70656,  71680)  16 x 64 col gates
//   scr  : [71680,  80384)  early: R/C/prefix sums; late: 544x16 scaled conv weights
//   st   : [80384,  80640)  per-bg scalars
//   meta : [80640,  81184)  544 packed k-decode entries (as int)
//   zpad : [81184,  81216)  guaranteed-zero words for branchless OOB loads
#define L_WMAP  65536
#define L_AH    69632
#define L_AW    70656
#define L_SCR   71680
#define L_ST    80384
#define L_META  80640
#define L_ZPAD  81184
#define L_TOTAL 81216

__global__ __launch_bounds__(1024, 1)
void ema_fused(const float* __restrict__ x,
               const float* __restrict__ w1, const float* __restrict__ b1,
               const float* __restrict__ w3, const float* __restrict__ b3,
               const float* __restrict__ w5, const float* __restrict__ b5,
               const float* __restrict__ gnw, const float* __restrict__ gnb,
               const float* __restrict__ ecw, const float* __restrict__ ecb,
               float* __restrict__ out)
{
  extern __shared__ float smem[];
  float* simg = smem;
  float* wmap = smem + L_WMAP;
  float* a_h  = smem + L_AH;
  float* a_w  = smem + L_AW;
  float* scr  = smem + L_SCR;
  float* st   = smem + L_ST;

  const int t  = threadIdx.x;
  const int bg = blockIdx.x;
  const float* gx   = x   + (size_t)bg * 65536;
  float*       gout = out + (size_t)bg * 65536;

  const unsigned lds_base = (unsigned)(uintptr_t)(void*)simg;   // low 32 bits = LDS offset

  // ---- Phase 1: async DMA of the whole 256KB group image into LDS ----
  {
    const float4* src = (const float4*)gx;
    for (int i = t; i < 16384; i += 1024) {
      unsigned loff = lds_base + (unsigned)i * 16u;
      const float4* g = src + i;
      asm volatile("global_load_async_to_lds_b128 %0, %1, off"
                   :: "v"(loff), "v"(g) : "memory");
    }
    asm volatile("s_wait_asynccnt 0x0" ::: "memory");
  }
  __syncthreads();

  // ---- Phase 2: row sums R (scr[0:1024]) and col sums C (scr[1024:2048]) ----
  {
    const int ci = t >> 6, r = t & 63;
    const float* row = simg + ci * 4096 + r * 64;
    float s = 0.f;
    for (int j = 0; j < 64; ++j) s += row[(j + t) & 63];
    scr[t] = s;
    const float* colb = simg + ci * 4096 + r;
    float c = 0.f;
    for (int j = 0; j < 64; ++j) c += colb[((j + t) & 63) * 64];
    scr[1024 + t] = c;
  }
  __syncthreads();

  // ---- Phase 3: 1x1 conv over channels on [x_h ; x_w], sigmoid gates ----
  {
    const int co = t & 15, pos = t >> 4;   // 16 co x 64 pos = 1024
    float sh = b1[co], sw = b1[co];
    for (int ci = 0; ci < 16; ++ci) {
      float wgt = w1[co * 16 + ci];
      sh += wgt * scr[ci * 64 + pos] * (1.f / 64.f);
      sw += wgt * scr[1024 + ci * 64 + pos] * (1.f / 64.f);
    }
    a_h[co * 64 + pos] = sigmoidf_(sh);
    a_w[co * 64 + pos] = sigmoidf_(sw);
  }
  __syncthreads();

  // ---- Phase 4: GN stats of gated = x * a_h * a_w ----
  {
    const int ci = t >> 6, r = t & 63;
    const float ah = a_h[ci * 64 + r];
    const float* row = simg + ci * 4096 + r * 64;
    float s1 = 0.f, s2 = 0.f;
    for (int j = 0; j < 64; ++j) {
      int xx = (j + t) & 63;
      float v = row[xx] * a_w[ci * 64 + xx];
      s1 += v; s2 += v * v;
    }
    wmap[t]        = s1 * ah;        // red1
    wmap[1024 + t] = s2 * ah * ah;   // red2
  }
  __syncthreads();
  if (t < 16) {
    float s1 = 0.f, s2 = 0.f;
    for (int j = 0; j < 64; ++j) { s1 += wmap[t * 64 + j]; s2 += wmap[1024 + t * 64 + j]; }
    float mu  = s1 * (1.f / 4096.f);
    float var = s2 * (1.f / 4096.f) - mu * mu;
    st[t]      = mu;
    st[16 + t] = rsqrtf(var + GN_EPS);
  }
  __syncthreads();

  // ---- Phase 5a: prefix sums of R and C per channel ----
  if (t < 16) {
    float accR = 0.f, accC = 0.f;
    float* PR = scr + 2048 + t * 65;
    float* PC = scr + 3088 + t * 65;
    PR[0] = 0.f; PC[0] = 0.f;
    for (int j = 0; j < 64; ++j) {
      accR += scr[t * 64 + j];
      accC += scr[1024 + t * 64 + j];
      PR[j + 1] = accR; PC[j + 1] = accC;
    }
  }
  __syncthreads();

  // ---- Phase 5b: exact per-channel conv means via inclusion-exclusion ----
  if (t < 16) {
    const int co = t;
    for (int which = 0; which < 2; ++which) {
      const int ks   = which ? 5 : 3;
      const int rad  = which ? 2 : 1;
      const int ntap = ks * ks;
      const float* wk = which ? w5 : w3;
      float acc = 4096.f * (which ? b5[co] : b3[co]);
      for (int ci = 0; ci < 16; ++ci) {
        const float* PR = scr + 2048 + ci * 65;
        const float* PC = scr + 3088 + ci * 65;
        const float total = PR[64];
        for (int tap = 0; tap < ntap; ++tap) {
          int dy = tap / ks - rad, dx = tap % ks - rad;
          int er0 = 1, er1 = 0, ec0 = 1, ec1 = 0;
          if (dy < 0) { er0 = 64 + dy; er1 = 63; } else if (dy > 0) { er0 = 0; er1 = dy - 1; }
          if (dx < 0) { ec0 = 64 + dx; ec1 = 63; } else if (dx > 0) { ec0 = 0; ec1 = dx - 1; }
          float rect = total;
          if (er1 >= er0) rect -= PR[er1 + 1] - PR[er0];
          if (ec1 >= ec0) rect -= PC[ec1 + 1] - PC[ec0];
          if (er1 >= er0 && ec1 >= ec0)
            for (int rr = er0; rr <= er1; ++rr)
              for (int cc = ec0; cc <= ec1; ++cc)
                rect += simg[ci * 4096 + rr * 64 + cc];
          acc += wk[(co * 16 + ci) * ntap + tap] * rect;
        }
      }
      st[(which ? 116 : 100) + co] = acc * (1.f / 4096.f);
    }
  }
  __syncthreads();

  // ---- Phase 5c: softmaxes, ECA, per-bg coefficients (serial, tiny) ----
  if (t == 0) {
    float m2[16], m3[16], gb[16], x11[16], x21[16], x31[16];
    for (int c = 0; c < 16; ++c) { m2[c] = st[100 + c]; m3[c] = st[116 + c]; gb[c] = gnb[c]; }
    auto softmax16 = [&](const float* v, float* o) {
      float mx = v[0];
      for (int c = 1; c < 16; ++c) mx = fmaxf(mx, v[c]);
      float s = 0.f;
      for (int c = 0; c < 16; ++c) { o[c] = __expf(v[c] - mx); s += o[c]; }
      float inv = 1.f / s;
      for (int c = 0; c < 16; ++c) o[c] *= inv;
    };
    softmax16(gb, x11); softmax16(m2, x21); softmax16(m3, x31);
    const float e0 = ecw[0], e1 = ecw[1], e2 = ecw[2], eb = ecb[0];
    auto eca = [&](const float* m, int c) {
      float l = (c > 0)  ? m[c - 1] : 0.f;
      float r = (c < 15) ? m[c + 1] : 0.f;
      return sigmoidf_(e0 * l + e1 * m[c] + e2 * r + eb);
    };
    float K0 = 0.f;
    for (int c = 0; c < 16; ++c) {
      float ca = eca(gb, c) + eca(m2, c) + eca(m3, c);
      float cA = x21[c] + x31[c];
      float cB = x11[c] + x31[c];
      float cC = x11[c] + x21[c];
      float rs = st[16 + c], mu = st[c], gw = gnw[c];
      st[32 + c] = ca;             // per-channel output scale
      st[48 + c] = cB;             // 3x3 collapse coef
      st[64 + c] = cC;             // 5x5 collapse coef
      st[80 + c] = cA * rs * gw;   // kA for gated term
      K0 += cA * (gb[c] - mu * rs * gw) + cB * b3[c] + cC * b5[c];
    }
    st[96] = K0;
  }
  __syncthreads();

  // ---- Phase 5d: bake kA into a_h; build scaled B matrix + k-decode meta ----
  {
    const int ci = t >> 6, r = t & 63;
    a_h[ci * 64 + r] *= st[80 + ci];
  }
  for (int idx = t; idx < 544 * 16; idx += 1024) {
    int k = idx >> 4, co = idx & 15;
    float v;
    if (k < 144) { int tap = k >> 4, ci = k & 15; v = st[48 + co] * w3[(co * 16 + ci) * 9 + tap]; }
    else { int kk = k - 144; int tap = kk >> 4, ci = kk & 15; v = st[64 + co] * w5[(co * 16 + ci) * 25 + tap]; }
    scr[k * 16 + co] = v;
  }
  if (t < 544) {
    int k = t;
    int tap, ci, dy, dx;
    if (k < 144) { tap = k >> 4; ci = k & 15; dy = tap / 3 - 1; dx = tap % 3 - 1; }
    else { int kk = k - 144; tap = kk >> 4; ci = kk & 15; dy = tap / 5 - 2; dx = tap % 5 - 2; }
    int off = ci * 4096 + dy * 64 + dx + 130;               // >= 0, < 2^17
    ((int*)(smem + L_META))[k] = off | ((dy + 2) << 20) | ((dx + 2) << 23);
  }
  if (t < 32) smem[L_ZPAD + t] = 0.f;                        // zero pad for OOB selects
  __syncthreads();

  // ---- Phase 6: fused implicit-GEMM conv via V_WMMA_F32_16X16X4_F32 ----
  // Tile: M = 16 x-positions of one row, N = 16 out-channels, K = 544 (tap,ci).
  // kb outer / 4 tiles inner: decode + B-fragment amortized over 4 WMMAs.
  {
    const int lane = t & 31, wv = t >> 5;
    const int half = lane >> 4;    // K-pair selector per A/B layout
    const int m    = lane & 15;    // M for A, N for B
    const int* metat = (const int*)(smem + L_META);
    for (int batch = 0; batch < 2; ++batch) {
      int ty[4], tx[4], tb[4];
#pragma unroll
      for (int i = 0; i < 4; ++i) {
        int tile = wv * 8 + batch * 4 + i;
        ty[i] = tile >> 2;
        tx[i] = (tile & 3) << 4;
        tb[i] = ty[i] * 64 + tx[i];
      }
      v8f zero = {};
      v8f acc[4];
#pragma unroll
      for (int i = 0; i < 4; ++i) acc[i] = zero;

      for (int kb = 0; kb < 544; kb += 4) {
        const int k0 = kb + half * 2;
        const int me0 = metat[k0], me1 = metat[k0 + 1];
        v2f b;
        b.x = scr[k0 * 16 + m];
        b.y = scr[k0 * 16 + 16 + m];
        const int dy0 = ((me0 >> 20) & 7) - 2;
        const int dy1 = ((me1 >> 20) & 7) - 2;
        const int xo0 = m + (((me0 >> 23) & 7) - 2);
        const int xo1 = m + (((me1 >> 23) & 7) - 2);
        const int ao0 = (me0 & 0xFFFFF) - 130 + m;
        const int ao1 = (me1 & 0xFFFFF) - 130 + m;
#pragma unroll
        for (int i = 0; i < 4; ++i) {
          bool ok0 = ((unsigned)(ty[i] + dy0) < 64u) && ((unsigned)(tx[i] + xo0) < 64u);
          bool ok1 = ((unsigned)(ty[i] + dy1) < 64u) && ((unsigned)(tx[i] + xo1) < 64u);
          int i0 = ok0 ? (ao0 + tb[i]) : L_ZPAD;   // select address, not value
          int i1 = ok1 ? (ao1 + tb[i]) : L_ZPAD;
          v2f a;
          a.x = smem[i0];
          a.y = smem[i1];
          acc[i] = __builtin_amdgcn_wmma_f32_16x16x4_f32(false, a, false, b, (short)0,
                                                         acc[i], false, false);
        }
      }
      // Row-sum D over its 16 N-lanes: staged butterfly so all 8 permutes per
      // stage are independent (one dscnt wait per stage, not per value).
#pragma unroll
      for (int i = 0; i < 4; ++i) {
        float s[8];
#pragma unroll
        for (int v = 0; v < 8; ++v) s[v] = acc[i][v];
#pragma unroll
        for (int d = 1; d <= 8; d <<= 1) {
#pragma unroll
          for (int v = 0; v < 8; ++v) s[v] += __shfl_xor(s[v], d, 32);
        }
        if (m == 0) {
#pragma unroll
          for (int v = 0; v < 8; ++v) wmap[tb[i] + half * 8 + v] = s[v];
        }
      }
    }
  }
  __syncthreads();

  // ---- Phase 7: add gated (x1) term, sigmoid ----
  for (int p = t; p < 4096; p += 1024) {
    int y = p >> 6, xc = p & 63;
    float g = 0.f;
#pragma unroll
    for (int ci = 0; ci < 16; ++ci)
      g += a_h[ci * 64 + y] * a_w[ci * 64 + xc] * simg[ci * 4096 + p];
    wmap[p] = sigmoidf_(wmap[p] + g + st[96]);
  }
  __syncthreads();

  // ---- Phase 8: scale in place in LDS, then async store LDS -> global ----
  {
    float4* sv = (float4*)simg;
    for (int i = t; i < 16384; i += 1024) {
      int e  = i << 2;
      int ci = e >> 12;
      int p  = e & 4095;
      float ca = st[32 + ci];
      float4 v = sv[i];
      v.x *= ca * wmap[p];
      v.y *= ca * wmap[p + 1];
      v.z *= ca * wmap[p + 2];
      v.w *= ca * wmap[p + 3];
      sv[i] = v;                               // in-place: each lane owns its float4
    }
    asm volatile("s_wait_dscnt 0x0" ::: "memory");   // own LDS writes visible to async engine
    const float4* gdst = (const float4*)gout;
    for (int i = t; i < 16384; i += 1024) {
      unsigned loff = lds_base + (unsigned)i * 16u;
      const float4* g = gdst + i;
      asm volatile("global_store_async_from_lds_b128 %0, %1, off"
                   :: "v"(g), "v"(loff) : "memory");
    }
    asm volatile("s_wait_asynccnt 0x0" ::: "memory");
  }
}

extern "C" void kernel_launch(void* const* d_in, const int* in_sizes, int n_in,
                              void* d_out, int out_size, void* d_ws, size_t ws_size,
                              hipStream_t stream) {
  (void)in_sizes; (void)n_in; (void)out_size; (void)d_ws; (void)ws_size;
  const float* x   = (const float*)d_in[0];
  const float* w1  = (const float*)d_in[1];
  const float* b1  = (const float*)d_in[2];
  const float* w3  = (const float*)d_in[3];
  const float* b3  = (const float*)d_in[4];
  const float* w5  = (const float*)d_in[5];
  const float* b5  = (const float*)d_in[6];
  const float* gnw = (const float*)d_in[7];
  const float* gnb = (const float*)d_in[8];
  const float* ecw = (const float*)d_in[9];
  const float* ecb = (const float*)d_in[10];
  float* out = (float*)d_out;
  const size_t shmem = (size_t)L_TOTAL * sizeof(float);   // 324,864 B < 320 KB WGP LDS
  ema_fused<<<512, 1024, shmem, stream>>>(x, w1, b1, w3, b3, w5, b5,
                                          gnw, gnb, ecw, ecb, out);
}